// ESM_ensamble_cls_env2_52458730553833
// MI455X (gfx1250) — compile-verified
//
#include <hip/hip_runtime.h>
#include <hip/hip_bf16.h>

typedef float v2f __attribute__((ext_vector_type(2)));
typedef float v8f __attribute__((ext_vector_type(8)));

#define Bc 8
#define Lc 512
#define Hc 20
#define Dc 1280
#define Kc 32

// ---------------------------------------------------------------------------
// Kernel 1: per (batch, rep) — pooled attention weights at row pos[b],
// iterative top-32 argmax, gather + mean of selected rep rows, and the
// CLS row copy. One block per (b, r): 16 blocks x 256 threads.
// ---------------------------------------------------------------------------
__global__ __launch_bounds__(256) void prep_kernel(
    const float* __restrict__ rep1, const float* __restrict__ rep2,
    const float* __restrict__ attn1, const float* __restrict__ attn2,
    const int* __restrict__ pos,
    float* __restrict__ Xcls, float* __restrict__ Xenv)
{
    const int b = blockIdx.x & 7;
    const int r = blockIdx.x >> 3;
    const float* rep  = r ? rep2  : rep1;
    const float* attn = r ? attn2 : attn1;
    const int p = pos[b];
    const int t = threadIdx.x;

    __shared__ float wbuf[Lc];
    __shared__ float smax[256];
    __shared__ int   sidx[256];
    __shared__ int   sel[Kc];

    // pooled attention weight: w[l] = mean_h attn[b,h,p,l]
    for (int l = t; l < Lc; l += 256) {
        float s = 0.f;
        const float* ap = attn + ((size_t)b * Hc) * Lc * Lc + (size_t)p * Lc + l;
        #pragma unroll 4
        for (int h = 0; h < Hc; ++h) s += ap[(size_t)h * Lc * Lc];
        wbuf[l] = s * (1.0f / (float)Hc);
    }

    // CLS branch row: rep[b, 0, :]
    const int rowbase = (r * 8 + b) * Dc;
    for (int d = t; d < Dc; d += 256)
        Xcls[rowbase + d] = rep[(size_t)b * Lc * Dc + d];
    __syncthreads();

    // top-K = 32 by iterative argmax (tie -> lowest index, matches top_k)
    for (int it = 0; it < Kc; ++it) {
        float v0 = wbuf[t], v1 = wbuf[t + 256];
        float v;  int idx;
        if (v1 > v0) { v = v1; idx = t + 256; } else { v = v0; idx = t; }
        smax[t] = v; sidx[t] = idx;
        __syncthreads();
        for (int off = 128; off > 0; off >>= 1) {
            if (t < off) {
                float ov = smax[t + off]; int oi = sidx[t + off];
                if (ov > smax[t] || (ov == smax[t] && oi < sidx[t])) {
                    smax[t] = ov; sidx[t] = oi;
                }
            }
            __syncthreads();
        }
        if (t == 0) { sel[it] = sidx[0]; wbuf[sidx[0]] = -3.402823466e+38f; }
        __syncthreads();
    }

    // mean of selected rows (mean-before-projection: exact by linearity)
    for (int d = t; d < Dc; d += 256) {
        float acc = 0.f;
        #pragma unroll 4
        for (int k = 0; k < Kc; ++k)
            acc += rep[((size_t)b * Lc + sel[k]) * Dc + d];
        Xenv[rowbase + d] = acc * (1.0f / (float)Kc);
    }
}

// ---------------------------------------------------------------------------
// Kernel 2: Y = X[16 x 1280] @ W[1280 x 1280] for both (cls, env) matrices
// using V_WMMA_F32_16X16X4_F32. One wave per 16-wide output tile:
// 80 tiles per matrix, 160 waves = 20 blocks x 8 waves.
// ---------------------------------------------------------------------------
__global__ __launch_bounds__(256) void wmma_gemm_kernel(
    const float* __restrict__ Xcls, const float* __restrict__ Wcls, float* __restrict__ Ycls,
    const float* __restrict__ Xenv, const float* __restrict__ Wenv, float* __restrict__ Yenv)
{
    const int wave = threadIdx.x >> 5;
    const int lane = threadIdx.x & 31;
    int tile = blockIdx.x * 8 + wave;      // 0..159, grid sized exactly

    const float* X; const float* W; float* Y;
    if (tile < 80) { X = Xcls; W = Wcls; Y = Ycls; }
    else           { X = Xenv; W = Wenv; Y = Yenv; tile -= 80; }

    const int n0   = tile * 16;
    const int m    = lane & 15;     // M index (A) / N index (B,C)
    const int half = lane >> 4;     // half-wave select
    const int kb   = half * 2;      // K sub-offset held by this lane

    v8f c = {};
#if __has_builtin(__builtin_amdgcn_wmma_f32_16x16x4_f32)
    for (int k0 = 0; k0 < Dc; k0 += 4) {
        v2f a, b;
        // A 16x4: lanes 0-15 hold M=lane K={0,1}; lanes 16-31 hold K={2,3}
        a.x = X[m * Dc + k0 + kb];
        a.y = X[m * Dc + k0 + kb + 1];
        // B 4x16: lanes 0-15 hold K rows {0,1} at N=lane; lanes 16-31 rows {2,3}
        b.x = W[(size_t)(k0 + kb)     * Dc + n0 + m];
        b.y = W[(size_t)(k0 + kb + 1) * Dc + n0 + m];
        c = __builtin_amdgcn_wmma_f32_16x16x4_f32(
                /*neg_a=*/false, a, /*neg_b=*/false, b,
                /*c_mod=*/(short)0, c, /*reuse_a=*/false, /*reuse_b=*/false);
    }
#else
    // scalar fallback (should not be hit on gfx1250)
    for (int i = 0; i < 8; ++i) {
        float acc = 0.f;
        for (int k = 0; k < Dc; ++k)
            acc += X[(i + half * 8) * Dc + k] * W[(size_t)k * Dc + n0 + m];
        c[i] = acc;
    }
#endif
    // C/D 16x16: VGPR i -> M = i + half*8, N = lane&15
    #pragma unroll
    for (int i = 0; i < 8; ++i)
        Y[(size_t)(i + half * 8) * Dc + n0 + m] = c[i];
}

// ---------------------------------------------------------------------------
// Kernel 3: fold biases + per-feature scales + classifier vectors into the
// scalar logits. One block per batch element.
// ---------------------------------------------------------------------------
__global__ __launch_bounds__(256) void reduce_kernel(
    const float* __restrict__ Ycls, const float* __restrict__ Yenv,
    const float* __restrict__ cls_b, const float* __restrict__ env_b,
    const float* __restrict__ clsclf_W, const float* __restrict__ clsclf_b,
    const float* __restrict__ envclf_W, const float* __restrict__ envclf_b,
    const float* __restrict__ cc1, const float* __restrict__ cc2,
    const float* __restrict__ ec1, const float* __restrict__ ec2,
    float* __restrict__ out)
{
    const int b = blockIdx.x;
    const int t = threadIdx.x;
    __shared__ float sred[256];

    float s = 0.f;
    for (int d = t; d < Dc; d += 256) {
        float y1 = Ycls[(size_t)b * Dc + d]       + cls_b[d];
        float y2 = Ycls[(size_t)(8 + b) * Dc + d] + cls_b[d];
        s += (cc1[d] * y1 + cc2[d] * y2) * clsclf_W[d];
        float f1 = Yenv[(size_t)b * Dc + d]       + env_b[d];
        float f2 = Yenv[(size_t)(8 + b) * Dc + d] + env_b[d];
        s += (ec1[d] * f1 + ec2[d] * f2) * envclf_W[d];
    }
    sred[t] = s;
    __syncthreads();
    for (int off = 128; off > 0; off >>= 1) {
        if (t < off) sred[t] += sred[t + off];
        __syncthreads();
    }
    if (t == 0) out[b] = 0.5f * (sred[0] + clsclf_b[0] + envclf_b[0]);
}

extern "C" void kernel_launch(void* const* d_in, const int* in_sizes, int n_in,
                              void* d_out, int out_size, void* d_ws, size_t ws_size,
                              hipStream_t stream) {
    const float* rep1     = (const float*)d_in[0];
    const float* rep2     = (const float*)d_in[1];
    const float* attn1    = (const float*)d_in[2];
    const float* attn2    = (const float*)d_in[3];
    const int*   pos      = (const int*)  d_in[4];
    const float* cls_W    = (const float*)d_in[5];
    const float* cls_b    = (const float*)d_in[6];
    const float* env_W    = (const float*)d_in[7];
    const float* env_b    = (const float*)d_in[8];
    const float* clsclf_W = (const float*)d_in[9];
    const float* clsclf_b = (const float*)d_in[10];
    const float* envclf_W = (const float*)d_in[11];
    const float* envclf_b = (const float*)d_in[12];
    const float* cls_c1   = (const float*)d_in[13];
    const float* cls_c2   = (const float*)d_in[14];
    const float* env_c1   = (const float*)d_in[15];
    const float* env_c2   = (const float*)d_in[16];

    float* ws   = (float*)d_ws;
    float* Xcls = ws;                    // [16,1280]
    float* Xenv = ws + 16 * Dc;          // [16,1280]
    float* Ycls = ws + 2 * 16 * Dc;      // [16,1280]
    float* Yenv = ws + 3 * 16 * Dc;      // [16,1280]

    prep_kernel<<<16, 256, 0, stream>>>(rep1, rep2, attn1, attn2, pos, Xcls, Xenv);
    wmma_gemm_kernel<<<20, 256, 0, stream>>>(Xcls, cls_W, Ycls, Xenv, env_W, Yenv);
    reduce_kernel<<<8, 256, 0, stream>>>(Ycls, Yenv, cls_b, env_b,
                                         clsclf_W, clsclf_b, envclf_W, envclf_b,
                                         cls_c1, cls_c2, env_c1, env_c2,
                                         (float*)d_out);
}